// SVDPlusPlusNet_412316861026
// MI455X (gfx1250) — compile-verified
//
#include <hip/hip_runtime.h>

// CDNA5 / gfx1250 — wave32, WMMA f32 16x16x4.
typedef __attribute__((ext_vector_type(2))) float v2f;
typedef __attribute__((ext_vector_type(8))) float v8f;

#define LATENT 64
#define ROWS_PER_WAVE 16

__global__ __launch_bounds__(256) void svdpp_score_wmma_f32(
    const int*   __restrict__ user_ids,
    const int*   __restrict__ item_ids,
    const float* __restrict__ user_emb,
    const float* __restrict__ item_emb,
    const float* __restrict__ user_bias,
    const float* __restrict__ item_bias,
    const float* __restrict__ global_bias,
    const float* __restrict__ implicit_cache,
    float*       __restrict__ out,
    int B)
{
    const int lane         = threadIdx.x & 31;
    const int waveInBlock  = threadIdx.x >> 5;
    const int wavesPerBlk  = blockDim.x >> 5;
    const int tile         = blockIdx.x * wavesPerBlk + waveInBlock;
    const int base         = tile * ROWS_PER_WAVE;

    // A-matrix 16x4 f32 layout (ISA 7.12.2): lane L holds row M = L%16,
    // columns K = 2*(L/16) .. 2*(L/16)+1 in its two VGPRs. The B operand
    // (column N = L%16) needs the identical per-lane addressing from the
    // item row, so every lane does three 8-byte loads per K-step.
    const int mrow = lane & 15;
    const int half = lane >> 4;

    int row = base + mrow;
    row = (row < B) ? row : (B - 1);          // clamp: keep EXEC all-ones

    const int uid = user_ids[row];
    const int iid = item_ids[row];

    const float* uePtr = user_emb       + (size_t)uid * LATENT;
    const float* imPtr = implicit_cache + (size_t)uid * LATENT;
    const float* iePtr = item_emb       + (size_t)iid * LATENT;

    v8f acc0 = {};
    v8f acc1 = {};

    // 16 x V_WMMA_F32_16X16X4_F32, two accumulators to break the RAW chain.
    #pragma unroll
    for (int k = 0; k < LATENT; k += 4) {
        const int c = k + 2 * half;
        v2f ue = *(const v2f*)(uePtr + c);
        v2f im = *(const v2f*)(imPtr + c);
        v2f ie = *(const v2f*)(iePtr + c);
        v2f a  = ue + im;
        if (((k >> 2) & 1) == 0) {
            acc0 = __builtin_amdgcn_wmma_f32_16x16x4_f32(
                /*neg_a=*/false, a, /*neg_b=*/false, ie,
                /*c_mod=*/(short)0, acc0, /*reuse_a=*/false, /*reuse_b=*/false);
        } else {
            acc1 = __builtin_amdgcn_wmma_f32_16x16x4_f32(
                false, a, false, ie, (short)0, acc1, false, false);
        }
    }

    // D layout (16x16 f32 C/D): lanes 0-15 hold M=j (VGPR j), N=lane;
    // lanes 16-31 hold M=j+8, N=lane-16.  Diagonal D[m][m]:
    //   m in 0..7  -> lane m      , acc[m]
    //   m in 8..15 -> lane m + 16 , acc[m-8]
    const int idx = lane & 7;
    float d0 = acc0[0];
    float d1 = acc1[0];
    #pragma unroll
    for (int j = 1; j < 8; ++j) {
        d0 = (idx == j) ? acc0[j] : d0;
        d1 = (idx == j) ? acc1[j] : d1;
    }
    const float dot = d0 + d1;

    const bool writer = (lane < 8) || (lane >= 24);
    const int  m      = (lane < 16) ? lane : (lane - 16);
    const int  orow   = base + m;
    if (writer && orow < B) {
        const float gb = global_bias[0];
        const float ub = user_bias[uid];   // [NUM_USERS,1] -> flat index uid
        const float ib = item_bias[iid];   // [NUM_ITEMS,1] -> flat index iid
        out[orow] = gb + ub + ib + dot;
    }
}

extern "C" void kernel_launch(void* const* d_in, const int* in_sizes, int n_in,
                              void* d_out, int out_size, void* d_ws, size_t ws_size,
                              hipStream_t stream) {
    const int*   user_ids       = (const int*)  d_in[0];
    const int*   item_ids       = (const int*)  d_in[1];
    const float* user_emb       = (const float*)d_in[2];
    const float* item_emb       = (const float*)d_in[3];
    const float* user_bias      = (const float*)d_in[4];
    const float* item_bias      = (const float*)d_in[5];
    const float* global_bias    = (const float*)d_in[6];
    const float* implicit_cache = (const float*)d_in[7];
    float*       out            = (float*)d_out;

    const int B = in_sizes[0];

    const int threads      = 256;                      // 8 waves
    const int rowsPerBlock = (threads / 32) * ROWS_PER_WAVE;  // 128
    const int blocks       = (B + rowsPerBlock - 1) / rowsPerBlock;

    svdpp_score_wmma_f32<<<blocks, threads, 0, stream>>>(
        user_ids, item_ids, user_emb, item_emb,
        user_bias, item_bias, global_bias, implicit_cache,
        out, B);
}